// Fit_90031104459546
// MI455X (gfx1250) — compile-verified
//
#include <hip/hip_runtime.h>

typedef __attribute__((ext_vector_type(2))) float v2f;
typedef __attribute__((ext_vector_type(8))) float v8f;

#define NPTS 8192
#define SQRT2_C 1.4142f
#define EPS_C 1e-12f
#define WS_STRIDE 32   // floats per batch in workspace: [0..5]=s1,cx1,cy1,s2,cx2,cy2  [8..16]=f_hat

// ---------------------------------------------------------------------------
// Kernel 1: per-batch normalization statistics (centroid xy + mean xy dist)
// ---------------------------------------------------------------------------
__device__ __forceinline__ float blockReduce256(float v, float* red, int t) {
  red[t] = v; __syncthreads();
  #pragma unroll
  for (int s = 128; s > 0; s >>= 1) {
    if (t < s) red[t] += red[t + s];
    __syncthreads();
  }
  float r = red[0];
  __syncthreads();
  return r;
}

__global__ void __launch_bounds__(256)
stats_kernel(const float* __restrict__ pts1, const float* __restrict__ pts2,
             float* __restrict__ ws) {
  const int b = blockIdx.x;
  const int t = threadIdx.x;
  const float* p1 = pts1 + (size_t)b * NPTS * 3;
  const float* p2 = pts2 + (size_t)b * NPTS * 3;
  __shared__ float red[256];

  float sx1 = 0.f, sy1 = 0.f, sx2 = 0.f, sy2 = 0.f;
  for (int p = t; p < NPTS; p += 256) {
    sx1 += p1[p * 3 + 0]; sy1 += p1[p * 3 + 1];
    sx2 += p2[p * 3 + 0]; sy2 += p2[p * 3 + 1];
  }
  sx1 = blockReduce256(sx1, red, t);
  sy1 = blockReduce256(sy1, red, t);
  sx2 = blockReduce256(sx2, red, t);
  sy2 = blockReduce256(sy2, red, t);
  const float inN = 1.0f / (float)NPTS;
  const float cx1 = sx1 * inN, cy1 = sy1 * inN;
  const float cx2 = sx2 * inN, cy2 = sy2 * inN;

  float d1 = 0.f, d2 = 0.f;
  for (int p = t; p < NPTS; p += 256) {
    float ax = p1[p * 3 + 0] - cx1, ay = p1[p * 3 + 1] - cy1;
    float bx = p2[p * 3 + 0] - cx2, by = p2[p * 3 + 1] - cy2;
    d1 += sqrtf(ax * ax + ay * ay);
    d2 += sqrtf(bx * bx + by * by);
  }
  d1 = blockReduce256(d1, red, t);
  d2 = blockReduce256(d2, red, t);

  if (t == 0) {
    float* wsb = ws + b * WS_STRIDE;
    wsb[0] = SQRT2_C / (d1 * inN);  // s1
    wsb[1] = cx1; wsb[2] = cy1;
    wsb[3] = SQRT2_C / (d2 * inN);  // s2
    wsb[4] = cx2; wsb[5] = cy2;
  }
}

// ---------------------------------------------------------------------------
// Kernel 2: Gram matrix via V_WMMA_F32_16X16X4_F32, 9x9 eigen, 3x3 rank-2,
//           final T2^T * F_rank2 * T1  -> out[b*9..]
// ---------------------------------------------------------------------------
__global__ void __launch_bounds__(128)
gram_kernel(const float* __restrict__ pts1, const float* __restrict__ pts2,
            const float* __restrict__ wts, float* __restrict__ ws,
            float* __restrict__ out) {
  const int b = blockIdx.x;
  const int t = threadIdx.x;
  const int wid = t >> 5;
  const int lane = t & 31;
  const float* p1 = pts1 + (size_t)b * NPTS * 3;
  const float* p2 = pts2 + (size_t)b * NPTS * 3;
  const float* wb = wts + (size_t)b * NPTS;
  float* wsb = ws + b * WS_STRIDE;

  const float s1 = wsb[0], cx1 = wsb[1], cy1 = wsb[2];
  const float s2 = wsb[3], cx2 = wsb[4], cy2 = wsb[5];

  __shared__ float stage[4][32 * 16];  // per-wave staging of 32 padded rows
  __shared__ float gsum[4][16 * 16];   // per-wave 16x16 Gram accumulators
  __shared__ float A9[81];
  __shared__ float Vsh[81];

  v8f acc = {0.f, 0.f, 0.f, 0.f, 0.f, 0.f, 0.f, 0.f};

  const int m  = lane & 15;        // component index (A) / column (B)
  const int kb = (lane >> 4) << 1; // K base: 0 for lanes 0-15, 2 for 16-31
  const int base = wid * (NPTS / 4);

  for (int it = 0; it < (NPTS / 4) / 32; ++it) {
    const int p = base + it * 32 + lane;
    float x1 = p1[p * 3 + 0], y1 = p1[p * 3 + 1], z1 = p1[p * 3 + 2];
    float x2 = p2[p * 3 + 0], y2 = p2[p * 3 + 1], z2 = p2[p * 3 + 2];
    float w  = wb[p];
    float X1 = s1 * (x1 - cx1 * z1), Y1 = s1 * (y1 - cy1 * z1), Z1 = z1;
    float X2 = s2 * (x2 - cx2 * z2), Y2 = s2 * (y2 - cy2 * z2);
    float n1 = X1 * X1 + Y1 * Y1 + Z1 * Z1;
    float n2 = X2 * X2 + Y2 * Y2 + 1.0f;
    float inv = w / fmaxf(sqrtf(n1 * n2), EPS_C);
    float* row = &stage[wid][lane * 16];
    row[0] = inv * X2 * X1; row[1] = inv * X2 * Y1; row[2] = inv * X2 * Z1;
    row[3] = inv * Y2 * X1; row[4] = inv * Y2 * Y1; row[5] = inv * Y2 * Z1;
    row[6] = inv * X1;      row[7] = inv * Y1;      row[8] = inv * Z1;
    #pragma unroll
    for (int c = 9; c < 16; ++c) row[c] = 0.f;
    __syncthreads();
    #pragma unroll
    for (int j = 0; j < 8; ++j) {
      const int r0 = j * 4 + kb;
      v2f a;
      a.x = stage[wid][r0 * 16 + m];
      a.y = stage[wid][(r0 + 1) * 16 + m];
      // A (16x4) and B (4x16) lane layouts coincide for our symmetric use:
      // lane l holds component l%16 of points 2*(l/16), 2*(l/16)+1.
      acc = __builtin_amdgcn_wmma_f32_16x16x4_f32(false, a, false, a,
                                                  (short)0, acc, false, false);
    }
    __syncthreads();
  }

  {
    const int rowoff = (lane >> 4) * 8;
    #pragma unroll
    for (int r = 0; r < 8; ++r)
      gsum[wid][(rowoff + r) * 16 + (lane & 15)] = acc[r];
  }
  __syncthreads();

  if (t < 81) {
    int i = t / 9, j = t % 9;
    A9[t] = gsum[0][i * 16 + j] + gsum[1][i * 16 + j] +
            gsum[2][i * 16 + j] + gsum[3][i * 16 + j];
    Vsh[t] = (i == j) ? 1.f : 0.f;
  }
  __syncthreads();

  // Parallel cyclic Jacobi on 9x9 symmetric A9 (lanes 0..8 do rows)
  for (int sweep = 0; sweep < 8; ++sweep) {
    for (int p = 0; p < 8; ++p) {
      for (int q = p + 1; q < 9; ++q) {
        float app = A9[p * 9 + p];
        float aqq = A9[q * 9 + q];
        float apq = A9[p * 9 + q];
        float c = 1.f, s = 0.f;
        if (fabsf(apq) > 1e-22f) {
          float theta = 0.5f * (aqq - app) / apq;
          float tt = 1.0f / (fabsf(theta) + sqrtf(theta * theta + 1.0f));
          if (theta < 0.f) tt = -tt;
          c = rsqrtf(tt * tt + 1.0f);
          s = tt * c;
        }
        float okp = 0.f, okq = 0.f, vkp = 0.f, vkq = 0.f;
        if (t < 9) {
          okp = A9[t * 9 + p]; okq = A9[t * 9 + q];
          vkp = Vsh[t * 9 + p]; vkq = Vsh[t * 9 + q];
        }
        __syncthreads();
        if (t < 9) {
          float nkp, nkq;
          if (t == p)      { nkp = c * c * app - 2.f * c * s * apq + s * s * aqq; nkq = 0.f; }
          else if (t == q) { nkp = 0.f; nkq = s * s * app + 2.f * c * s * apq + c * c * aqq; }
          else             { nkp = c * okp - s * okq; nkq = s * okp + c * okq; }
          A9[t * 9 + p] = nkp; A9[p * 9 + t] = nkp;
          A9[t * 9 + q] = nkq; A9[q * 9 + t] = nkq;
          Vsh[t * 9 + p] = c * vkp - s * vkq;
          Vsh[t * 9 + q] = s * vkp + c * vkq;
        }
        __syncthreads();
      }
    }
  }

  if (t == 0) {
    // smallest eigenvector of XtX = last right singular vector of X
    int jm = 0; float mv = A9[0];
    for (int j = 1; j < 9; ++j) { float d = A9[j * 9 + j]; if (d < mv) { mv = d; jm = j; } }
    float f[9]; float nn = 0.f;
    for (int i = 0; i < 9; ++i) { f[i] = Vsh[i * 9 + jm]; nn += f[i] * f[i]; }
    float innf = rsqrtf(fmaxf(nn, 1e-30f));
    for (int i = 0; i < 9; ++i) { f[i] *= innf; wsb[8 + i] = f[i]; }

    // 3x3: M = F^T F, smallest eigenvector v3 -> F_rank2 = F (I - v3 v3^T)
    float M[3][3];
    M[0][0] = f[0]*f[0]+f[3]*f[3]+f[6]*f[6];
    M[0][1] = f[0]*f[1]+f[3]*f[4]+f[6]*f[7];
    M[0][2] = f[0]*f[2]+f[3]*f[5]+f[6]*f[8];
    M[1][1] = f[1]*f[1]+f[4]*f[4]+f[7]*f[7];
    M[1][2] = f[1]*f[2]+f[4]*f[5]+f[7]*f[8];
    M[2][2] = f[2]*f[2]+f[5]*f[5]+f[8]*f[8];
    M[1][0] = M[0][1]; M[2][0] = M[0][2]; M[2][1] = M[1][2];
    float V3[3][3] = {{1.f,0.f,0.f},{0.f,1.f,0.f},{0.f,0.f,1.f}};
    for (int sweep = 0; sweep < 6; ++sweep) {
      for (int pp = 0; pp < 2; ++pp) {
        for (int qq = pp + 1; qq < 3; ++qq) {
          float apq = M[pp][qq];
          if (fabsf(apq) < 1e-24f) continue;
          float theta = 0.5f * (M[qq][qq] - M[pp][pp]) / apq;
          float tt = 1.0f / (fabsf(theta) + sqrtf(theta * theta + 1.0f));
          if (theta < 0.f) tt = -tt;
          float c = rsqrtf(tt * tt + 1.0f), s = tt * c;
          for (int k = 0; k < 3; ++k) {
            float akp = M[k][pp], akq = M[k][qq];
            M[k][pp] = c * akp - s * akq; M[k][qq] = s * akp + c * akq;
          }
          for (int k = 0; k < 3; ++k) {
            float apk = M[pp][k], aqk = M[qq][k];
            M[pp][k] = c * apk - s * aqk; M[qq][k] = s * apk + c * aqk;
          }
          for (int k = 0; k < 3; ++k) {
            float vkp = V3[k][pp], vkq = V3[k][qq];
            V3[k][pp] = c * vkp - s * vkq; V3[k][qq] = s * vkp + c * vkq;
          }
        }
      }
    }
    int j3 = 0; float m3 = M[0][0];
    if (M[1][1] < m3) { m3 = M[1][1]; j3 = 1; }
    if (M[2][2] < m3) { m3 = M[2][2]; j3 = 2; }
    float v0 = V3[0][j3], v1 = V3[1][j3], v2 = V3[2][j3];
    float Fv0 = f[0]*v0 + f[1]*v1 + f[2]*v2;
    float Fv1 = f[3]*v0 + f[4]*v1 + f[5]*v2;
    float Fv2 = f[6]*v0 + f[7]*v1 + f[8]*v2;
    float Fr[3][3];
    Fr[0][0]=f[0]-Fv0*v0; Fr[0][1]=f[1]-Fv0*v1; Fr[0][2]=f[2]-Fv0*v2;
    Fr[1][0]=f[3]-Fv1*v0; Fr[1][1]=f[4]-Fv1*v1; Fr[1][2]=f[5]-Fv1*v2;
    Fr[2][0]=f[6]-Fv2*v0; Fr[2][1]=f[7]-Fv2*v1; Fr[2][2]=f[8]-Fv2*v2;

    // out = T2^T * Fr * T1
    float G[3][3];
    for (int j = 0; j < 3; ++j) {
      G[j][0] = Fr[j][0] * s1;
      G[j][1] = Fr[j][1] * s1;
      G[j][2] = Fr[j][2] - s1 * (cx1 * Fr[j][0] + cy1 * Fr[j][1]);
    }
    float* ob = out + b * 9;
    for (int l = 0; l < 3; ++l) {
      ob[0 * 3 + l] = s2 * G[0][l];
      ob[1 * 3 + l] = s2 * G[1][l];
      ob[2 * 3 + l] = G[2][l] - s2 * (cx2 * G[0][l] + cy2 * G[1][l]);
    }
  }
}

// ---------------------------------------------------------------------------
// Kernel 3: residual = X @ f_hat   (recompute rows, dot with f)
// ---------------------------------------------------------------------------
__global__ void __launch_bounds__(256)
residual_kernel(const float* __restrict__ pts1, const float* __restrict__ pts2,
                const float* __restrict__ wts, const float* __restrict__ ws,
                float* __restrict__ res) {
  const int idx = blockIdx.x * 256 + threadIdx.x;
  const int b = idx >> 13;          // N = 8192
  const int n = idx & (NPTS - 1);
  const float* wsb = ws + b * WS_STRIDE;
  const float s1 = wsb[0], cx1 = wsb[1], cy1 = wsb[2];
  const float s2 = wsb[3], cx2 = wsb[4], cy2 = wsb[5];
  const float f0 = wsb[8],  f1 = wsb[9],  f2 = wsb[10];
  const float f3 = wsb[11], f4 = wsb[12], f5 = wsb[13];
  const float f6 = wsb[14], f7 = wsb[15], f8 = wsb[16];

  const float* p1 = pts1 + ((size_t)b * NPTS + n) * 3;
  const float* p2 = pts2 + ((size_t)b * NPTS + n) * 3;
  float x1 = p1[0], y1 = p1[1], z1 = p1[2];
  float x2 = p2[0], y2 = p2[1], z2 = p2[2];
  float w  = wts[(size_t)b * NPTS + n];
  float X1 = s1 * (x1 - cx1 * z1), Y1 = s1 * (y1 - cy1 * z1), Z1 = z1;
  float X2 = s2 * (x2 - cx2 * z2), Y2 = s2 * (y2 - cy2 * z2);
  float n1 = X1 * X1 + Y1 * Y1 + Z1 * Z1;
  float n2 = X2 * X2 + Y2 * Y2 + 1.0f;
  float inv = w / fmaxf(sqrtf(n1 * n2), EPS_C);
  float r = inv * (X2 * (f0 * X1 + f1 * Y1 + f2 * Z1) +
                   Y2 * (f3 * X1 + f4 * Y1 + f5 * Z1) +
                          f6 * X1 + f7 * Y1 + f8 * Z1);
  res[idx] = r;
}

extern "C" void kernel_launch(void* const* d_in, const int* in_sizes, int n_in,
                              void* d_out, int out_size, void* d_ws, size_t ws_size,
                              hipStream_t stream) {
  const float* pts1 = (const float*)d_in[0];
  const float* pts2 = (const float*)d_in[1];
  const float* wts  = (const float*)d_in[2];
  float* out = (float*)d_out;
  float* ws  = (float*)d_ws;
  const int B = 256;

  stats_kernel<<<B, 256, 0, stream>>>(pts1, pts2, ws);
  gram_kernel<<<B, 128, 0, stream>>>(pts1, pts2, wts, ws, out);
  residual_kernel<<<(B * NPTS) / 256, 256, 0, stream>>>(pts1, pts2, wts, ws,
                                                        out + B * 9);
}